// SpGraphAttentionLayer_14078902796511
// MI455X (gfx1250) — compile-verified
//
#include <hip/hip_runtime.h>
#include <hip/hip_bf16.h>

typedef __attribute__((ext_vector_type(2))) float v2f;
typedef __attribute__((ext_vector_type(8))) float v8f;

#define IN_F   128
#define OUT_F  64
#define ALPHA  0.2f

// ---------------------------------------------------------------------------
// Kernel 1: h = x @ W^T + b  via V_WMMA_F32_16X16X4_F32 (full fp32 matrix pipe)
// One wave32 computes one 16x16 tile of H (N x 64). K-loop: 32 WMMAs of K=4.
// A (16x4 f32): lane(0..15)=row M, vgpr0=K0,vgpr1=K1; lanes 16..31: K2,K3.
// B (4x16 f32): same striping with columns of W^T == rows of W.
// C/D (16x16 f32): c[j] <-> (M = j + 8*(lane>=16), N = lane&15).
// ---------------------------------------------------------------------------
__global__ __launch_bounds__(256) void gat_gemm_wmma(
    const float* __restrict__ X, const float* __restrict__ W,
    const float* __restrict__ B, float* __restrict__ H, int nrows)
{
    const int wave = threadIdx.x >> 5;            // 8 waves / block
    const int lane = threadIdx.x & 31;
    const int tile = blockIdx.x * 8 + wave;       // flat 16x16 tile id
    const int tm   = (tile >> 2) * 16;            // row base (OUT_F/16 = 4 col tiles)
    const int tn   = (tile & 3) * 16;             // col base
    if (tm >= nrows) return;                      // wave-uniform: EXEC stays all-ones

    const int r  = lane & 15;
    const int hi = lane >> 4;                     // K sub-pair select
    const float* __restrict__ xrow = X + (size_t)(tm + r) * IN_F;
    const float* __restrict__ wrow = W + (size_t)(tn + r) * IN_F;

    v8f c = {};
    #pragma unroll
    for (int k = 0; k < IN_F; k += 4) {
        v2f a, b;
        a.x = xrow[k + 2 * hi + 0];
        a.y = xrow[k + 2 * hi + 1];
        b.x = wrow[k + 2 * hi + 0];
        b.y = wrow[k + 2 * hi + 1];
        c = __builtin_amdgcn_wmma_f32_16x16x4_f32(
                /*neg_a=*/false, a, /*neg_b=*/false, b,
                /*c_mod=*/(short)0, c, /*reuse_a=*/false, /*reuse_b=*/false);
    }

    const float bias = B[tn + r];
    #pragma unroll
    for (int j = 0; j < 8; ++j) {
        const int m = tm + j + 8 * hi;
        H[(size_t)m * OUT_F + tn + r] = c[j] + bias;
    }
}

// ---------------------------------------------------------------------------
// Kernel 2: f1[i] = h[i,:]·a1 ; f2[i] = h[i,:]·a2   (one thread per row)
// ---------------------------------------------------------------------------
__global__ __launch_bounds__(256) void gat_fvec(
    const float* __restrict__ H, const float* __restrict__ A,
    float* __restrict__ F1, float* __restrict__ F2, int n)
{
    const int i = blockIdx.x * blockDim.x + threadIdx.x;
    if (i >= n) return;
    const float* __restrict__ hr = H + (size_t)i * OUT_F;
    float s1 = 0.f, s2 = 0.f;
    #pragma unroll 8
    for (int j = 0; j < OUT_F; ++j) {
        const float h = hr[j];
        s1 = fmaf(h, A[j], s1);
        s2 = fmaf(h, A[OUT_F + j], s2);
    }
    F1[i] = s1;
    F2[i] = s2;
}

// ---------------------------------------------------------------------------
// Kernel 3: edge scatter. One wave32 per edge, float2 per lane (64 features).
// Lane 0 computes edge_e = exp(leaky_relu(f1[src]+f2[dst])) once, broadcasts
// via __shfl; all lanes do GLOBAL_ATOMIC_ADD_F32 into the accumulator.
// ---------------------------------------------------------------------------
__global__ __launch_bounds__(256) void gat_edge(
    const long long* __restrict__ adj, const float* __restrict__ H,
    const float* __restrict__ F1, const float* __restrict__ F2,
    float* __restrict__ OUT, float* __restrict__ RS, int nedges)
{
    const int lane = threadIdx.x & 31;
    const int e    = (int)((blockIdx.x * 256u + threadIdx.x) >> 5);
    if (e >= nedges) return;

    const long long s = adj[2 * (long long)e + 0];
    const long long d = adj[2 * (long long)e + 1];

    float ee = 0.f;
    if (lane == 0) {
        float v = F1[s] + F2[d];
        v  = (v > 0.f) ? v : ALPHA * v;
        ee = expf(v);
        atomicAdd(&RS[s], ee);
    }
    ee = __shfl(ee, 0, 32);

    const float2 hv = ((const float2*)(H + (size_t)d * OUT_F))[lane];
    float* dst = OUT + (size_t)s * OUT_F + 2 * lane;
    atomicAdd(dst + 0, ee * hv.x);
    atomicAdd(dst + 1, ee * hv.y);
}

// ---------------------------------------------------------------------------
// Kernel 4: out = elu(out / rowsum)   (elementwise)
// ---------------------------------------------------------------------------
__global__ __launch_bounds__(256) void gat_final(
    float* __restrict__ OUT, const float* __restrict__ RS, int n)
{
    const int idx = blockIdx.x * blockDim.x + threadIdx.x;
    if (idx >= n * OUT_F) return;
    const float v = OUT[idx] / RS[idx >> 6];
    OUT[idx] = (v > 0.f) ? v : (expf(v) - 1.f);
}

// ---------------------------------------------------------------------------
extern "C" void kernel_launch(void* const* d_in, const int* in_sizes, int n_in,
                              void* d_out, int out_size, void* d_ws, size_t ws_size,
                              hipStream_t stream)
{
    const float*     x   = (const float*)d_in[0];      // (N, 128)
    const float*     W   = (const float*)d_in[1];      // (64, 128)
    const float*     b   = (const float*)d_in[2];      // (64,)
    const float*     a   = (const float*)d_in[3];      // (128,)
    const long long* adj = (const long long*)d_in[4];  // (E, 2) int64

    const int N = in_sizes[0] / IN_F;
    const int E = in_sizes[4] / 2;

    // workspace layout
    float* H  = (float*)d_ws;                 // N * 64
    float* F1 = H  + (size_t)N * OUT_F;       // N
    float* F2 = F1 + N;                       // N
    float* RS = F2 + N;                       // N

    float* OUT = (float*)d_out;               // N * 64 accumulator -> result

    hipMemsetAsync(OUT, 0, (size_t)N * OUT_F * sizeof(float), stream);
    hipMemsetAsync(RS,  0, (size_t)N * sizeof(float), stream);

    // GEMM: one wave per 16x16 tile, 8 waves per block
    const int tilesM = (N + 15) / 16;
    const int tiles  = tilesM * (OUT_F / 16);
    gat_gemm_wmma<<<(tiles + 7) / 8, 256, 0, stream>>>(x, W, b, H, N);

    gat_fvec<<<(N + 255) / 256, 256, 0, stream>>>(H, a, F1, F2, N);

    // edge scatter: 1 wave per edge -> 8 edges per 256-thread block
    gat_edge<<<(E + 7) / 8, 256, 0, stream>>>(adj, H, F1, F2, OUT, RS, E);

    gat_final<<<((N * OUT_F) + 255) / 256, 256, 0, stream>>>(OUT, RS, N);
}